// FraudGNN_56633438765476
// MI455X (gfx1250) — compile-verified
//
#include <hip/hip_runtime.h>
#include <hip/hip_bf16.h>
#include <math.h>

typedef __attribute__((ext_vector_type(2))) float v2f;
typedef __attribute__((ext_vector_type(8))) float v8f;

#define THREADS 256

__device__ __forceinline__ void atomic_add_f32_nr(float* p, float v) {
    // relaxed, agent-scope, no-return -> global_atomic_add_f32 (STOREcnt)
    __hip_atomic_fetch_add(p, v, __ATOMIC_RELAXED, __HIP_MEMORY_SCOPE_AGENT);
}

// K0: every node has exactly one self-loop -> start degree at 1.
__global__ __launch_bounds__(THREADS) void k_init_deg(float* __restrict__ deg, int n) {
    int i = blockIdx.x * THREADS + threadIdx.x;
    if (i < n) deg[i] = 1.0f;
}

// K1: degree count over dst (L2-resident 4MB region, fire-and-forget atomics)
__global__ __launch_bounds__(THREADS) void k_deg(const int* __restrict__ dst,
                                                 float* __restrict__ deg, int e) {
    int i = blockIdx.x * THREADS + threadIdx.x;
    if (i < e) atomic_add_f32_nr(&deg[dst[i]], 1.0f);
}

// K2: dinv = deg^{-1/2} in place (deg >= 1 always, matches reference's guarded rsqrt)
__global__ __launch_bounds__(THREADS) void k_dinv(float* __restrict__ deg, int n) {
    int i = blockIdx.x * THREADS + threadIdx.x;
    if (i < n) deg[i] = rsqrtf(deg[i]);
}

// K3: layer-1 dense transform via V_WMMA_F32_16X16X4_F32.
//   A (16x4): 16 node rows of x, K=2..3 zero-padded.
//     layout: vgpr j, lanes 0-15 -> K=j, lanes 16-31 -> K=2+j; M = lane%16
//   B (4x16): W1 (2x8) zero-padded.
//     layout: vgpr j, lanes 0-15 -> row K=j, lanes 16-31 -> row K=2+j; N = lane%16
//   D (16x16): vgpr r, lanes 0-15 -> M=r, lanes 16-31 -> M=r+8; N = lane%16
// Writes hs[m][f] = h*dinv[m]  and seeds agg1[m][f] = h*dinv[m]^2 (self-loop).
__global__ __launch_bounds__(THREADS) void k_layer1(const float* __restrict__ x,
                                                    const float* __restrict__ W1,
                                                    const float* __restrict__ dinv,
                                                    float* __restrict__ hs,
                                                    float* __restrict__ agg1, int n) {
    int wave = (blockIdx.x * THREADS + threadIdx.x) >> 5;
    int lane = threadIdx.x & 31;
    int base = wave * 16;
    if (base >= n) return;                  // wave-uniform: EXEC all-ones for WMMA
    const bool fullTile = (base + 16 <= n); // wave-uniform fast path
    int half = lane >> 4;                   // 0 -> K=0,1 ; 1 -> K=2,3 (zero pad)
    int lm   = lane & 15;
    int m    = base + lm;

    // Branchless operand build: always-valid clamped loads + cndmask selects.
    int mc = m < n ? m : n - 1;
    float x0 = x[2 * (size_t)mc + 0];
    float x1 = x[2 * (size_t)mc + 1];
    bool aval = (half == 0) && (m < n);
    v2f a; a.x = aval ? x0 : 0.0f; a.y = aval ? x1 : 0.0f;

    int wi = lm & 7;                        // W1 is 16 floats: rows 0/1 x cols 0..7
    float w0 = W1[wi];
    float w1 = W1[8 + wi];
    bool bval = (half == 0) && (lm < 8);
    v2f b; b.x = bval ? w0 : 0.0f; b.y = bval ? w1 : 0.0f;

    v8f c = {};
    c = __builtin_amdgcn_wmma_f32_16x16x4_f32(false, a, false, b, (short)0, c, false, false);

    if (lm < 8) {                           // feature column f = lm
        int f = lm;
        int rowbase = base + 8 * half;
        if (fullTile) {
            #pragma unroll
            for (int r = 0; r < 8; ++r) {
                int mm   = rowbase + r;
                float di = dinv[mm];
                float v  = c[r];
                hs[(size_t)mm * 8 + f]   = v * di;
                agg1[(size_t)mm * 8 + f] = v * di * di;
            }
        } else {
            #pragma unroll
            for (int r = 0; r < 8; ++r) {
                int mm = rowbase + r;
                if (mm < n) {
                    float di = dinv[mm];
                    float v  = c[r];
                    hs[(size_t)mm * 8 + f]   = v * di;
                    agg1[(size_t)mm * 8 + f] = v * di * di;
                }
            }
        }
    }
}

// K4: layer-1 edge scatter: agg1[dst] += hs[src] * dinv[dst]  (8 feats)
__global__ __launch_bounds__(THREADS) void k_edge1(const int* __restrict__ src,
                                                   const int* __restrict__ dst,
                                                   const float* __restrict__ dinv,
                                                   const float* __restrict__ hs,
                                                   float* __restrict__ agg1, int e) {
    int i = blockIdx.x * THREADS + threadIdx.x;
    if (i >= e) return;
    int s = src[i], d = dst[i];
    float nd = dinv[d];
    const float4* hp = (const float4*)(hs + (size_t)s * 8);
    float4 h0 = hp[0];
    float4 h1 = hp[1];
    float* ap = agg1 + (size_t)d * 8;
    atomic_add_f32_nr(ap + 0, h0.x * nd);
    atomic_add_f32_nr(ap + 1, h0.y * nd);
    atomic_add_f32_nr(ap + 2, h0.z * nd);
    atomic_add_f32_nr(ap + 3, h0.w * nd);
    atomic_add_f32_nr(ap + 4, h1.x * nd);
    atomic_add_f32_nr(ap + 5, h1.y * nd);
    atomic_add_f32_nr(ap + 6, h1.z * nd);
    atomic_add_f32_nr(ap + 7, h1.w * nd);
}

// K5: layer-2 dense transform: t = relu(agg1 + b1) @ W2  (K=8 -> two chained K=4 WMMAs)
// hs2[m] = t*dinv[m]; seeds agg2[m] = t*dinv[m]^2 (self-loop).
__global__ __launch_bounds__(THREADS) void k_layer2(const float* __restrict__ agg1,
                                                    const float* __restrict__ b1,
                                                    const float* __restrict__ W2,
                                                    const float* __restrict__ dinv,
                                                    float* __restrict__ hs2,
                                                    float* __restrict__ agg2, int n) {
    int wave = (blockIdx.x * THREADS + threadIdx.x) >> 5;
    int lane = threadIdx.x & 31;
    int base = wave * 16;
    if (base >= n) return;
    const bool fullTile = (base + 16 <= n);
    int half = lane >> 4;
    int lm   = lane & 15;
    int m    = base + lm;
    bool mv  = (m < n);
    int mc   = mv ? m : n - 1;

    int k0 = 2 * half;                      // K for vgpr j=0 of WMMA #1 (WMMA #2: +4)
    size_t rb = (size_t)mc * 8;
    float g0 = agg1[rb + k0 + 0];
    float g1 = agg1[rb + k0 + 1];
    float g4 = agg1[rb + k0 + 4];
    float g5 = agg1[rb + k0 + 5];
    float c0 = b1[k0 + 0];
    float c1 = b1[k0 + 1];
    float c4 = b1[k0 + 4];
    float c5 = b1[k0 + 5];

    v2f a1, a2;
    a1.x = mv ? fmaxf(g0 + c0, 0.0f) : 0.0f;
    a1.y = mv ? fmaxf(g1 + c1, 0.0f) : 0.0f;
    a2.x = mv ? fmaxf(g4 + c4, 0.0f) : 0.0f;
    a2.y = mv ? fmaxf(g5 + c5, 0.0f) : 0.0f;

    float w0 = W2[k0 + 0];                  // W2 is 8 floats: always-valid loads
    float w1 = W2[k0 + 1];
    float w4 = W2[k0 + 4];
    float w5 = W2[k0 + 5];
    bool bval = (lm == 0);                  // only output column N==0 nonzero
    v2f b1v, b2v;
    b1v.x = bval ? w0 : 0.0f;
    b1v.y = bval ? w1 : 0.0f;
    b2v.x = bval ? w4 : 0.0f;
    b2v.y = bval ? w5 : 0.0f;

    v8f c = {};
    c = __builtin_amdgcn_wmma_f32_16x16x4_f32(false, a1, false, b1v, (short)0, c, false, false);
    c = __builtin_amdgcn_wmma_f32_16x16x4_f32(false, a2, false, b2v, (short)0, c, false, false);

    if (lm == 0) {                          // column N=0: lanes 0 (M=0..7), 16 (M=8..15)
        int rowbase = base + 8 * half;
        if (fullTile) {
            #pragma unroll
            for (int r = 0; r < 8; ++r) {
                int mm   = rowbase + r;
                float di = dinv[mm];
                float v  = c[r];
                hs2[mm]  = v * di;
                agg2[mm] = v * di * di;
            }
        } else {
            #pragma unroll
            for (int r = 0; r < 8; ++r) {
                int mm = rowbase + r;
                if (mm < n) {
                    float di = dinv[mm];
                    float v  = c[r];
                    hs2[mm]  = v * di;
                    agg2[mm] = v * di * di;
                }
            }
        }
    }
}

// K6: layer-2 edge scatter: agg2[dst] += hs2[src] * dinv[dst]
__global__ __launch_bounds__(THREADS) void k_edge2(const int* __restrict__ src,
                                                   const int* __restrict__ dst,
                                                   const float* __restrict__ dinv,
                                                   const float* __restrict__ hs2,
                                                   float* __restrict__ agg2, int e) {
    int i = blockIdx.x * THREADS + threadIdx.x;
    if (i >= e) return;
    int s = src[i], d = dst[i];
    atomic_add_f32_nr(&agg2[d], hs2[s] * dinv[d]);
}

// K7: out = sigmoid(agg2 + b2)
__global__ __launch_bounds__(THREADS) void k_final(const float* __restrict__ agg2,
                                                   const float* __restrict__ b2,
                                                   float* __restrict__ out, int n) {
    int i = blockIdx.x * THREADS + threadIdx.x;
    if (i < n) {
        float z = agg2[i] + b2[0];
        out[i] = 1.0f / (1.0f + expf(-z));
    }
}

extern "C" void kernel_launch(void* const* d_in, const int* in_sizes, int n_in,
                              void* d_out, int out_size, void* d_ws, size_t ws_size,
                              hipStream_t stream) {
    const float* x    = (const float*)d_in[0];
    const int*   edge = (const int*)d_in[1];
    const float* W1   = (const float*)d_in[2];
    const float* b1   = (const float*)d_in[3];
    const float* W2   = (const float*)d_in[4];
    const float* b2   = (const float*)d_in[5];

    const int N = in_sizes[0] / 2;        // x is [N, 2]
    const int E = in_sizes[1] / 2;        // edge_index is [2, E]
    const int* src = edge;
    const int* dst = edge + (size_t)E;

    // workspace layout (floats): [deg/dinv: N][hs: 8N (reused: hs2=N, agg2=N)][agg1: 8N]
    float* ws   = (float*)d_ws;
    float* deg  = ws;                      // becomes dinv in place
    float* hs   = ws + (size_t)N;
    float* agg1 = ws + (size_t)9 * N;
    float* hs2  = hs;                      // hs dead after K4 -> reuse
    float* agg2 = hs + (size_t)N;

    const int nBlkN = (N + THREADS - 1) / THREADS;
    const int nBlkE = (E + THREADS - 1) / THREADS;
    const int tiles = (N + 15) / 16;                       // 16 nodes per wave
    const int nBlkT = (tiles + (THREADS / 32) - 1) / (THREADS / 32);

    k_init_deg<<<nBlkN, THREADS, 0, stream>>>(deg, N);
    k_deg     <<<nBlkE, THREADS, 0, stream>>>(dst, deg, E);
    k_dinv    <<<nBlkN, THREADS, 0, stream>>>(deg, N);
    k_layer1  <<<nBlkT, THREADS, 0, stream>>>(x, W1, deg, hs, agg1, N);
    k_edge1   <<<nBlkE, THREADS, 0, stream>>>(src, dst, deg, hs, agg1, E);
    k_layer2  <<<nBlkT, THREADS, 0, stream>>>(agg1, b1, W2, deg, hs2, agg2, N);
    k_edge2   <<<nBlkE, THREADS, 0, stream>>>(src, dst, deg, hs2, agg2, E);
    k_final   <<<nBlkN, THREADS, 0, stream>>>(agg2, b2, (float*)d_out, N);
}